// GaussianRenderer_240518168822
// MI455X (gfx1250) — compile-verified
//
#include <hip/hip_runtime.h>
#include <math.h>

// ---------------------------------------------------------------------------
// Gaussian splat renderer for gfx1250 (MI455X).
//   H=W=512, N_GAUSS=4096, isotropic cov (25*I) -> alpha depends only on
//   squared screen distance. We express
//     q = log2( exp(-0.5*d^2/25) * opacity )
//       = A(gauss) . B(pixel),  K = 4:
//     A = [-2*C2*sx, -2*C2*sy, C2, C2*(sx^2+sy^2)+log2(op)]
//     B = [px, py, px^2+py^2, 1],  C2 = -0.5/25 * log2(e)
//   so alpha = min(exp2(q), 1). One V_WMMA_F32_16X16X4_F32 produces q for a
//   16-gaussian x 16-pixel tile (256 interactions / instruction, fp32 exact).
// Compute-bound: ~1.07e9 interactions, ~128 KB gaussian data (L0-resident),
// 5 MB output -> instruction count / latency hiding is everything.
// ---------------------------------------------------------------------------

#define IMG_H 512
#define IMG_W 512
#define NG    4096
#define NPIX  (IMG_H * IMG_W)

typedef __attribute__((ext_vector_type(2))) float v2f;
typedef __attribute__((ext_vector_type(8))) float v8f;

__device__ __forceinline__ float fast_exp2(float x) {
#if __has_builtin(__builtin_amdgcn_exp2f)
  return __builtin_amdgcn_exp2f(x);   // v_exp_f32, 1 trans op
#else
  return exp2f(x);
#endif
}

// Half-wave (xor-16) exchange. v_permlanex16_b32 with identity nibble selects
// makes lane i read opposite-row lane i: pure VALU, no LDS round-trip and no
// s_wait_dscnt on the serial transmittance chain (ds_bpermute cost ~25+ cyc).
__device__ __forceinline__ float xor16(float x) {
#if __has_builtin(__builtin_amdgcn_permlanex16)
  int i = __builtin_bit_cast(int, x);
  i = __builtin_amdgcn_permlanex16(i, i, 0x76543210, 0xFEDCBA98, false, false);
  return __builtin_bit_cast(float, i);
#else
  return __shfl_xor(x, 16, 32);
#endif
}

// ---------------------------------------------------------------------------
// Kernel 1: per-gaussian projection + feature packing (unsorted).
// ---------------------------------------------------------------------------
__global__ __launch_bounds__(256) void gs_preprocess(
    const float* __restrict__ pos, const float* __restrict__ col,
    const float* __restrict__ opac, const float* __restrict__ quat,
    const float* __restrict__ cam, const float* __restrict__ intr,
    float* __restrict__ key, float4* __restrict__ ufeat,
    float4* __restrict__ ucol) {
  int g = blockIdx.x * blockDim.x + threadIdx.x;
  if (g >= NG) return;

  float qw = quat[0], qx = quat[1], qy = quat[2], qz = quat[3];
  float qn = rsqrtf(qw * qw + qx * qx + qy * qy + qz * qz);
  qw *= qn; qx *= qn; qy *= qn; qz *= qn;
  float R00 = 1.f - 2.f * (qy * qy + qz * qz), R01 = 2.f * (qx * qy - qw * qz), R02 = 2.f * (qx * qz + qw * qy);
  float R10 = 2.f * (qx * qy + qw * qz), R11 = 1.f - 2.f * (qx * qx + qz * qz), R12 = 2.f * (qy * qz - qw * qx);
  float R20 = 2.f * (qx * qz - qw * qy), R21 = 2.f * (qy * qz + qw * qx), R22 = 1.f - 2.f * (qx * qx + qy * qy);

  float p0 = pos[3 * g + 0] - cam[0];
  float p1 = pos[3 * g + 1] - cam[1];
  float p2 = pos[3 * g + 2] - cam[2];
  float c0 = R00 * p0 + R01 * p1 + R02 * p2;
  float c1 = R10 * p0 + R11 * p1 + R12 * p2;
  float c2 = R20 * p0 + R21 * p1 + R22 * p2;

  float r0 = intr[0] * c0 + intr[1] * c1 + intr[2] * c2;
  float r1 = intr[3] * c0 + intr[4] * c1 + intr[5] * c2;
  float r2 = intr[6] * c0 + intr[7] * c1 + intr[8] * c2;
  float sx = r0 / r2, sy = r1 / r2;

  const float C2 = -0.02f * 1.4426950408889634f;  // -0.5/25 * log2(e)
  float op = opac[g];
  ufeat[g] = make_float4(-2.f * C2 * sx, -2.f * C2 * sy, C2,
                         C2 * (sx * sx + sy * sy) + log2f(op));
  ucol[g]  = make_float4(col[3 * g + 0], col[3 * g + 1], col[3 * g + 2], c2);
  key[g]   = r2;  // proj z (sort key)
}

// ---------------------------------------------------------------------------
// Kernel 2: single-block bitonic argsort (descending depth = argsort(-depth)),
// then gather packed arrays in sorted order. LDS: 32 KB of 320 KB.
// ---------------------------------------------------------------------------
__global__ __launch_bounds__(1024) void gs_sort_gather(
    const float* __restrict__ key_in, const float4* __restrict__ ufeat,
    const float4* __restrict__ ucol, float4* __restrict__ sfeat,
    float4* __restrict__ scol) {
  __shared__ float skey[NG];
  __shared__ int   sidx[NG];
  int t = threadIdx.x;
  for (int i = t; i < NG; i += 1024) { skey[i] = key_in[i]; sidx[i] = i; }
  __syncthreads();

  for (int k = 2; k <= NG; k <<= 1) {
    for (int j = k >> 1; j > 0; j >>= 1) {
      for (int i = t; i < NG; i += 1024) {
        int ixj = i ^ j;
        if (ixj > i) {
          float a = skey[i], b = skey[ixj];
          bool descSeg = ((i & k) == 0);
          bool doswap = descSeg ? (a < b) : (a > b);   // final order: descending
          if (doswap) {
            skey[i] = b; skey[ixj] = a;
            int ti = sidx[i]; sidx[i] = sidx[ixj]; sidx[ixj] = ti;
          }
        }
      }
      __syncthreads();
    }
  }
  for (int i = t; i < NG; i += 1024) {
    int o = sidx[i];
    sfeat[i] = ufeat[o];
    scol[i]  = ucol[o];
  }
}

// ---------------------------------------------------------------------------
// Kernel 3: render. One wave owns 16 pixels (one row segment).
// Per 16-gaussian chunk:
//   A frag (lane l<16: K0/K1 of gaussian l ; lane l>=16: K2/K3 of gaussian l-16)
//   B frag (lane n<16: (px,py) of pixel n  ; lane n>=16: (px^2+py^2, 1))
//   D: lane n    -> pixel n, D[v] = q(gaussian v,   pixel n)
//      lane n+16 -> pixel n, D[v] = q(gaussian v+8, pixel n)
// Each lane serially composites its 8 gaussians; the half-wave transmittance
// product is exchanged with one xor16 (v_permlanex16): T_next = T*P_lo*P_hi is
// identical on both halves. Accumulators (sum of T*alpha*chan) are order-
// independent partial sums, combined with xor16 at the end.
// ---------------------------------------------------------------------------
__global__ __launch_bounds__(256) void gs_render(
    const float4* __restrict__ sfeat, const float4* __restrict__ scol,
    const float* __restrict__ bg, float* __restrict__ out) {
  const int lane = threadIdx.x & 31;
  const int wave = threadIdx.x >> 5;
  const int half = lane >> 4;
  const int l15  = lane & 15;
  const int pix  = (blockIdx.x * 8 + wave) * 16 + l15;  // 16 px per wave, same row

  const float px = (float)(pix & (IMG_W - 1));
  const float py = (float)(pix >> 9);

  v2f bfrag;
  if (half == 0) { bfrag.x = px;                 bfrag.y = py;   }
  else           { bfrag.x = px * px + py * py;  bfrag.y = 1.0f; }

  float accR = 0.f, accG = 0.f, accB = 0.f, accD = 0.f, accA = 0.f;
  float T = 1.0f;
  const char* featb = (const char*)sfeat;

  // Unroll x2: cross-chunk dependence is only the scalar T chain, so the
  // scheduler can overlap chunk k+1's WMMA / v_exp_f32 / b128 loads with
  // chunk k's composite multiplies.
#pragma unroll 2
  for (int c = 0; c < NG / 16; ++c) {
    // next chunk's colors -> global_prefetch_b8 (harmless once L0-resident)
    __builtin_prefetch((const char*)(scol + (c + 1) * 16), 0, 1);

    // A fragment: 8B lane-dependent slice of the packed feature vector
    v2f a = *(const v2f*)(featb + (((c * 16 + l15) << 4) + (half << 3)));

    v8f cz = {};
    v8f q = __builtin_amdgcn_wmma_f32_16x16x4_f32(
        /*neg_a=*/false, a, /*neg_b=*/false, bfrag,
        /*c_mod=*/(short)0, cz, /*reuse_a=*/false, /*reuse_b=*/false);

    float alpha[8];
#pragma unroll
    for (int v = 0; v < 8; ++v) alpha[v] = fminf(fast_exp2(q[v]), 1.0f);

    float Pl = 1.0f;
#pragma unroll
    for (int v = 0; v < 8; ++v) Pl *= (1.0f - alpha[v]);
    float Po = xor16(Pl);                       // other half's product (VALU)

    float pf = half ? T * Po : T;               // running transmittance
    const float4* cd = scol + (c * 16 + half * 8);
#pragma unroll
    for (int v = 0; v < 8; ++v) {
      float w = pf * alpha[v];
      float4 ch = cd[v];
      accR = fmaf(w, ch.x, accR);
      accG = fmaf(w, ch.y, accG);
      accB = fmaf(w, ch.z, accB);
      accD = fmaf(w, ch.w, accD);
      accA += w;
      pf *= (1.0f - alpha[v]);
    }
    T *= Pl * Po;                               // identical on both halves
  }

  // combine the two half-wave partial sums for each pixel
  accR += xor16(accR);
  accG += xor16(accG);
  accB += xor16(accB);
  accD += xor16(accD);
  accA += xor16(accA);

  if (half == 0) {
    out[0 * NPIX + pix] = accR + bg[0];
    out[1 * NPIX + pix] = accG + bg[1];
    out[2 * NPIX + pix] = accB + bg[2];
    out[3 * NPIX + pix] = accD;
    out[4 * NPIX + pix] = accA;
  }
}

// ---------------------------------------------------------------------------
extern "C" void kernel_launch(void* const* d_in, const int* in_sizes, int n_in,
                              void* d_out, int out_size, void* d_ws, size_t ws_size,
                              hipStream_t stream) {
  const float* positions = (const float*)d_in[0];  // 4096*3
  const float* colors    = (const float*)d_in[1];  // 4096*3
  const float* opac      = (const float*)d_in[2];  // 4096
  const float* quat      = (const float*)d_in[3];  // 4
  const float* cam       = (const float*)d_in[4];  // 3
  const float* intr      = (const float*)d_in[5];  // 9
  const float* bg        = (const float*)d_in[6];  // 3

  char* ws = (char*)d_ws;
  float*  key   = (float*)(ws);                    // 16 KB
  float4* ufeat = (float4*)(ws + 16 * 1024);       // 64 KB
  float4* ucol  = (float4*)(ws + 80 * 1024);       // 64 KB
  float4* sfeat = (float4*)(ws + 144 * 1024);      // 64 KB
  float4* scol  = (float4*)(ws + 208 * 1024);      // 64 KB (272 KB total)

  gs_preprocess<<<NG / 256, 256, 0, stream>>>(positions, colors, opac, quat,
                                              cam, intr, key, ufeat, ucol);
  gs_sort_gather<<<1, 1024, 0, stream>>>(key, ufeat, ucol, sfeat, scol);
  gs_render<<<NPIX / 128, 256, 0, stream>>>(sfeat, scol, bg, (float*)d_out);
}